// SwitchSparceMLP_77094662964056
// MI455X (gfx1250) — compile-verified
//
#include <hip/hip_runtime.h>

#define Bb 4
#define Ss 2048
#define Dd 1024
#define Ff 4096
#define Ee 8
#define CAPc 320

typedef __attribute__((ext_vector_type(2))) float v2f;
typedef __attribute__((ext_vector_type(8))) float v8f;

// ---------------------------------------------------------------------------
// CDNA5 async global->LDS copy (16B per lane), tracked by ASYNCcnt.
// Inline asm per CDNA5 ISA + CDNA5 ISA (opcode 98).
// ---------------------------------------------------------------------------
__device__ __forceinline__ void async_copy_b128(const float* gsrc, float* lds_dst)
{
    unsigned lds_addr = (unsigned)(unsigned long long)lds_dst;      // low 32b = LDS offset
    unsigned long long gaddr = (unsigned long long)gsrc;
    asm volatile("global_load_async_to_lds_b128 %0, %1, off"
                 :: "v"(lds_addr), "v"(gaddr)
                 : "memory");
}
__device__ __forceinline__ void wait_async0()
{
    asm volatile("s_wait_asynccnt 0x0" ::: "memory");
}

// ---------------------------------------------------------------------------
// K1: router — logits, softmax max-prob, argmax (first occurrence)
// one wave32 per token
// ---------------------------------------------------------------------------
__global__ __launch_bounds__(256)
void router_kernel(const float* __restrict__ hs, const float* __restrict__ rw,
                   float* __restrict__ logits_out, int* __restrict__ expert_idx,
                   float* __restrict__ max_prob)
{
    int token = blockIdx.x * (blockDim.x >> 5) + (threadIdx.x >> 5);
    int lane  = threadIdx.x & 31;
    if (token >= Bb * Ss) return;

    const float* h = hs + (size_t)token * Dd;
    float acc[Ee];
#pragma unroll
    for (int e = 0; e < Ee; ++e) acc[e] = 0.0f;

    for (int d = lane; d < Dd; d += 32) {
        float hv = h[d];
        const float* r = rw + (size_t)d * Ee;
#pragma unroll
        for (int e = 0; e < Ee; ++e) acc[e] += hv * r[e];
    }
#pragma unroll
    for (int e = 0; e < Ee; ++e) {
        float v = acc[e];
#pragma unroll
        for (int off = 16; off >= 1; off >>= 1) v += __shfl_down(v, off, 32);
        acc[e] = v;
    }
    if (lane == 0) {
        float mx = acc[0]; int mi = 0;
#pragma unroll
        for (int e = 1; e < Ee; ++e) if (acc[e] > mx) { mx = acc[e]; mi = e; }
        float sum = 0.0f;
#pragma unroll
        for (int e = 0; e < Ee; ++e) sum += expf(acc[e] - mx);
#pragma unroll
        for (int e = 0; e < Ee; ++e) logits_out[(size_t)token * Ee + e] = acc[e];
        expert_idx[token] = mi;
        max_prob[token]   = 1.0f / sum;   // exp(0)/sum == softmax at argmax
    }
}

// ---------------------------------------------------------------------------
// K2: capacity scan (sequential over S, semantically required by cumsum)
// ---------------------------------------------------------------------------
__global__ __launch_bounds__(256)
void assign_kernel(const int* __restrict__ expert_idx,
                   int* __restrict__ dispatch, int* __restrict__ slot_used,
                   float* __restrict__ eidx_out)
{
    int b = blockIdx.x;
    for (int i = threadIdx.x; i < Ee * CAPc; i += blockDim.x) {
        dispatch[b * Ee * CAPc + i]  = 0;
        slot_used[b * Ee * CAPc + i] = 0;
    }
    __syncthreads();
    if (threadIdx.x == 0) {
        int cnt[Ee];
#pragma unroll
        for (int e = 0; e < Ee; ++e) cnt[e] = 0;
        for (int s = 0; s < Ss; ++s) {
            int e = expert_idx[b * Ss + s];
            int c = cnt[e]++;
            bool valid = (c < CAPc);
            if (valid) {
                dispatch[(b * Ee + e) * CAPc + c]  = s;
                slot_used[(b * Ee + e) * CAPc + c] = 1;
            }
            eidx_out[b * Ss + s] = valid ? (float)e : 0.0f;
        }
    }
}

// ---------------------------------------------------------------------------
// K3: out = max_prob * hidden for every token (baseline for unrouted/dropped)
// ---------------------------------------------------------------------------
__global__ __launch_bounds__(256)
void scale_init_kernel(const float* __restrict__ hs,
                       const float* __restrict__ max_prob,
                       float* __restrict__ out)
{
    size_t idx = (size_t)blockIdx.x * blockDim.x + threadIdx.x; // over B*S*(D/4)
    if (idx >= (size_t)Bb * Ss * (Dd / 4)) return;
    size_t tok = idx / (Dd / 4);
    float p = max_prob[tok];
    float4 h = ((const float4*)hs)[idx];
    float4 o; o.x = p * h.x; o.y = p * h.y; o.z = p * h.z; o.w = p * h.w;
    ((float4*)out)[idx] = o;
}

// ---------------------------------------------------------------------------
// K4/K5: grouped GEMM on the fp32 WMMA path (V_WMMA_F32_16X16X4_F32)
//   FFN1: hg[g] = relu( gather(hidden, dispatch[g]) @ wi[e] )   [320x1024]x[1024x4096]
//   FFN2: scatter( (hg[g] @ wo[e]) * max_prob , used slots )    [320x4096]x[4096x1024]
// Block tile 64x256, BK=16, 8 waves (2x4), wave tile 32x64.
// Double-buffered LDS, staged with GLOBAL_LOAD_ASYNC_TO_LDS_B128 (ASYNCcnt).
// ---------------------------------------------------------------------------
template <bool FFN1>
__global__ __launch_bounds__(256)
void moe_gemm_kernel(const float* __restrict__ Abase,   // hidden | hg
                     const float* __restrict__ Wbase,   // wi     | wo
                     const int*   __restrict__ dispatch,
                     const int*   __restrict__ slot_used,
                     const float* __restrict__ max_prob,
                     float*       __restrict__ Obase)   // hg     | out
{
    constexpr int BM = 64, BN = 256, BK = 16;
    constexpr int AST = 28;        // 112B rows: 16B aligned, r*28 mod 64 distinct for r=0..15
    constexpr int BST = BN + 8;    // 1056B rows: aligned, half-waves on disjoint banks
    constexpr int KDIM = FFN1 ? Dd : Ff;
    constexpr int NDIM = FFN1 ? Ff : Dd;

    __shared__ float As[2][BM][AST];
    __shared__ float Bs[2][BK][BST];
    __shared__ int   toks[BM];
    __shared__ int   usedv[BM];
    __shared__ float scales[BM];

    const int g  = blockIdx.z;           // 0..31 -> (b,e)
    const int b  = g / Ee;
    const int e  = g % Ee;
    const int m0 = blockIdx.y * BM;      // CAP = 5 * 64 exactly
    const int n0 = blockIdx.x * BN;

    const int tid   = threadIdx.x;
    const int wave  = tid >> 5;
    const int lane  = tid & 31;
    const int wm    = wave >> 2;         // 0..1
    const int wn    = wave & 3;          // 0..3
    const int lrow  = lane & 15;
    const int khalf = (lane >> 4) << 1;  // 0 or 2 (A/B K-half per ISA layout)

    for (int i = tid; i < BM; i += blockDim.x) {
        int c = m0 + i;
        int t = dispatch[g * CAPc + c];
        toks[i]   = t;
        usedv[i]  = slot_used[g * CAPc + c];
        scales[i] = max_prob[b * Ss + t];
    }
    __syncthreads();   // toks needed by the A-tile gather below

    v8f acc[2][4];
#pragma unroll
    for (int i = 0; i < 2; ++i)
#pragma unroll
        for (int j = 0; j < 4; ++j) { v8f z = {}; acc[i][j] = z; }

    const float* Wg = Wbase + (size_t)e * KDIM * NDIM;

    const int arow  = tid >> 2;          // 0..63
    const int acol  = (tid & 3) << 2;    // 0,4,8,12
    const int brow  = tid >> 4;          // 0..15
    const int bcol0 = (tid & 15) << 4;   // 0..240

    // async stage of one K-tile into LDS buffer `buf`
    auto stage = [&](int k0, int buf) {
        const float* aptr;
        if (FFN1)
            aptr = Abase + ((size_t)b * Ss + toks[arow]) * Dd + k0 + acol;
        else
            aptr = Abase + ((size_t)(g * CAPc + m0 + arow)) * Ff + k0 + acol;
        async_copy_b128(aptr, &As[buf][arow][acol]);                 // 64x16 tile
        const float* bptr = Wg + (size_t)(k0 + brow) * NDIM + n0 + bcol0;
#pragma unroll
        for (int j = 0; j < 4; ++j)                                  // 16x256 tile
            async_copy_b128(bptr + j * 4, &Bs[buf][brow][bcol0 + j * 4]);
    };

    constexpr int NK = KDIM / BK;
    stage(0, 0);
    wait_async0();
    __syncthreads();

    for (int kt = 0; kt < NK; ++kt) {
        const int cur = kt & 1;
        const int nxt = cur ^ 1;
        if (kt + 1 < NK) stage((kt + 1) * BK, nxt);   // overlap with compute

        // ---- 4 K-steps of 4 on the current staged tile ----
#pragma unroll
        for (int kk = 0; kk < BK; kk += 4) {
            v2f afrag[2], bfrag[4];
#pragma unroll
            for (int i = 0; i < 2; ++i) {
                int r = wm * 32 + i * 16 + lrow;
                afrag[i].x = As[cur][r][kk + khalf];
                afrag[i].y = As[cur][r][kk + khalf + 1];
            }
#pragma unroll
            for (int j = 0; j < 4; ++j) {
                int c = wn * 64 + j * 16 + lrow;
                bfrag[j].x = Bs[cur][kk + khalf][c];
                bfrag[j].y = Bs[cur][kk + khalf + 1][c];
            }
#pragma unroll
            for (int i = 0; i < 2; ++i)
#pragma unroll
                for (int j = 0; j < 4; ++j)
                    acc[i][j] = __builtin_amdgcn_wmma_f32_16x16x4_f32(
                        false, afrag[i], false, bfrag[j],
                        (short)0, acc[i][j], false, false);
        }

        if (kt + 1 < NK) wait_async0();   // next tile fully in LDS
        __syncthreads();                  // all waves done with `cur`, next tile visible
    }

    // ---- epilogue ----
#pragma unroll
    for (int i = 0; i < 2; ++i) {
#pragma unroll
        for (int j = 0; j < 4; ++j) {
#pragma unroll
            for (int v = 0; v < 8; ++v) {
                int mloc = wm * 32 + i * 16 + v + ((lane >> 4) << 3);
                int n    = n0 + wn * 64 + j * 16 + lrow;
                float val = acc[i][j][v];
                if (FFN1) {
                    val = fmaxf(val, 0.0f);
                    Obase[(size_t)(g * CAPc + m0 + mloc) * Ff + n] = val;
                } else {
                    if (usedv[mloc]) {
                        int tok = toks[mloc];
                        Obase[((size_t)b * Ss + tok) * Dd + n] = scales[mloc] * val;
                    }
                }
            }
        }
    }
}

// ---------------------------------------------------------------------------
extern "C" void kernel_launch(void* const* d_in, const int* in_sizes, int n_in,
                              void* d_out, int out_size, void* d_ws, size_t ws_size,
                              hipStream_t stream)
{
    const float* hs = (const float*)d_in[0];   // [B,S,D]
    const float* rw = (const float*)d_in[1];   // [D,E]
    const float* wi = (const float*)d_in[2];   // [E,D,F]
    const float* wo = (const float*)d_in[3];   // [E,F,D]

    float* out      = (float*)d_out;                      // [B,S,D]
    float* logits   = out + (size_t)Bb * Ss * Dd;         // [B,S,E]
    float* eidx_out = logits + (size_t)Bb * Ss * Ee;      // [B,S] (index as float)

    char* ws = (char*)d_ws;
    int*   expert_idx = (int*)ws;                                     // B*S
    float* max_prob   = (float*)(ws + (size_t)Bb * Ss * 4);           // B*S
    int*   dispatch   = (int*)(ws + (size_t)2 * Bb * Ss * 4);         // B*E*CAP
    int*   slot_used  = (int*)(ws + (size_t)2 * Bb * Ss * 4
                                  + (size_t)Bb * Ee * CAPc * 4);      // B*E*CAP
    float* hg         = (float*)(ws + (size_t)2 * Bb * Ss * 4
                                    + (size_t)2 * Bb * Ee * CAPc * 4);// B*E*CAP*F

    // K1: router
    router_kernel<<<(Bb * Ss) / 8, 256, 0, stream>>>(hs, rw, logits, expert_idx, max_prob);
    // K2: capacity scan / dispatch build
    assign_kernel<<<Bb, 256, 0, stream>>>(expert_idx, dispatch, slot_used, eidx_out);
    // K3: baseline out = max_prob * hidden
    scale_init_kernel<<<((Bb * Ss * (Dd / 4)) + 255) / 256, 256, 0, stream>>>(hs, max_prob, out);
    // K4: FFN1 (gather + GEMM + relu) -> hg
    dim3 g1(Ff / 256, CAPc / 64, Bb * Ee);
    moe_gemm_kernel<true><<<g1, 256, 0, stream>>>(hs, wi, dispatch, slot_used, max_prob, hg);
    // K5: FFN2 (GEMM + scaled scatter) -> out
    dim3 g2(Dd / 256, CAPc / 64, Bb * Ee);
    moe_gemm_kernel<false><<<g2, 256, 0, stream>>>(hg, wo, dispatch, slot_used, max_prob, out);
}